// DeepLabV3_10780367913762
// MI455X (gfx1250) — compile-verified
//
#include <hip/hip_runtime.h>

typedef __attribute__((ext_vector_type(2))) float v2f;
typedef __attribute__((ext_vector_type(8))) float v8f;
typedef __attribute__((ext_vector_type(4))) unsigned int u32x4;
typedef __attribute__((ext_vector_type(8))) int i32x8;
typedef __attribute__((ext_vector_type(4))) int i32x4;

namespace {
constexpr int CIN    = 256;
constexpr int WPX    = 128;   // pixels per row (W)
constexpr int HPX    = 128;
constexpr int BATCH  = 8;
constexpr int NCLS   = 19;
constexpr int NPROTO = 10;
constexpr int NPAD   = 192;   // 190 prototypes padded to 12 N-tiles
constexpr int PITCH  = 132;   // LDS pitch (dwords); TDM pad gives this for free
constexpr int MPITCH = 200;   // LDS pitch for masks [128][192]
constexpr int XTILE_OFF = CIN * PITCH + WPX;          // dwords: buf + pinv
constexpr int LDS_DW    = XTILE_OFF + CIN * PITCH;    // + xtile
constexpr int OUTHW  = 512;
}

#if __has_builtin(__builtin_amdgcn_wmma_f32_16x16x4_f32)
__device__ __forceinline__ v8f wmma_f32(v2f a, v2f b, v8f c) {
  // (neg_a, A, neg_b, B, c_mod, C, reuse_a, reuse_b)
  return __builtin_amdgcn_wmma_f32_16x16x4_f32(false, a, false, b, (short)0, c,
                                               false, false);
}
#else
__device__ __forceinline__ v8f wmma_f32(v2f a, v2f b, v8f c) {
  #pragma unroll
  for (int i = 0; i < 8; ++i) c[i] += a[0] * b[0] + a[1] * b[1];
  return c;
}
#endif

#if __has_builtin(__builtin_amdgcn_tensor_load_to_lds) && \
    __has_builtin(__builtin_amdgcn_s_wait_tensorcnt)
#define HAVE_TDM 1
#else
#define HAVE_TDM 0
#endif

// ---------------------------------------------------------------------------
// Kernel 1: l2-normalize prototypes [190][256] -> ws [192][256] (rows 190/191=0)
// ---------------------------------------------------------------------------
__global__ void proto_l2norm(const float* __restrict__ protos,
                             float* __restrict__ out) {
  const int j = threadIdx.x;
  if (j >= NPAD) return;
  if (j < NCLS * NPROTO) {
    float ss = 0.f;
    for (int d = 0; d < CIN; ++d) { float v = protos[j * CIN + d]; ss += v * v; }
    const float inv = 1.f / fmaxf(sqrtf(ss), 1e-12f);
    for (int d = 0; d < CIN; ++d) out[j * CIN + d] = protos[j * CIN + d] * inv;
  } else {
    for (int d = 0; d < CIN; ++d) out[j * CIN + d] = 0.f;
  }
}

// ---------------------------------------------------------------------------
// Kernel 2: fused head for one (b,h) image row. TDM DMAs the 128KB x-slab
// (256 ch x 128 px, stride 16384) into LDS with pad-to-pitch-132, then
// GEMM1+ReLU, GEMM2, L2(W), LN(256)+L2, prototype GEMM, max-over-10, LN(19).
// ---------------------------------------------------------------------------
__global__ void __launch_bounds__(256)
fused_head(const float* __restrict__ x,  const float* __restrict__ w1,
           const float* __restrict__ b1, const float* __restrict__ w2,
           const float* __restrict__ b2, const float* __restrict__ ln1g,
           const float* __restrict__ ln1b, const float* __restrict__ ln2g,
           const float* __restrict__ ln2b, const float* __restrict__ protosN,
           float* __restrict__ seg) {
  extern __shared__ float smem[];
  float* buf   = smem;                 // [CIN][PITCH] recycled through stages
  float* pinv  = smem + CIN * PITCH;   // [WPX] per-pixel 1/l2norm
  float* xtile = smem + XTILE_OFF;     // [CIN][PITCH] TDM-landed input slab

  const int tid  = threadIdx.x;
  const int lane = tid & 31;
  const int wv   = tid >> 5;            // wave 0..7 owns pixel tile wv
  const int bb   = blockIdx.x >> 7;     // batch
  const int hh   = blockIdx.x & (HPX - 1);
  const int m16  = lane & 15;
  const int hi   = (lane >= 16);
  const int klo  = hi ? 2 : 0;          // ISA: lanes<16 hold K+0/1, lanes>=16 K+2/3
  const int px0  = wv * 16;
  const int col  = px0 + m16;

  const float* xrow = x + ((size_t)bb * CIN) * (HPX * WPX) + (size_t)hh * WPX;

#if HAVE_TDM
  // ---- Tensor Data Mover: async 2D tile load global -> LDS ------------------
  if (wv == 0) {
    const unsigned long long ga = (unsigned long long)(const void*)xrow;
    const unsigned int ga_lo = (unsigned int)ga;
    const unsigned int ga_hi = (unsigned int)(ga >> 32);
    const unsigned int ldsa  = (unsigned int)(uintptr_t)xtile;  // low 32 = LDS byte addr
    // D# group0: count=1 | lds_addr | global_addr[56:0] | type=2
    u32x4 g0 = { 1u, ldsa, ga_lo, (ga_hi & 0x01FFFFFFu) | (2u << 30) };
    // D# group1: data_size=4B(code2), pad_enable, pad_interval=128dw(code6),
    // pad_amount=4dw(code3); tensor_dim0=128, tensor_dim1=256,
    // tile_dim0=128, tile_dim1=256, tensor_dim0_stride=16384
    i32x8 g1 = { (int)((2u << 16) | (1u << 20) | (6u << 22) | (3u << 25)),
                 (int)(128u << 16),   // tensor_dim0[15:0] in [31:16]
                 (int)(256u << 16),   // tensor_dim1[15:0] in [31:16]
                 (int)(128u << 16),   // tile_dim0 in [31:16]
                 256,                 // tile_dim1
                 16384,               // tensor_dim0_stride[31:0]
                 0, 0 };
    i32x4 gz4 = { 0, 0, 0, 0 };
    i32x8 gz8 = { 0, 0, 0, 0, 0, 0, 0, 0 };
    __builtin_amdgcn_tensor_load_to_lds(g0, g1, gz4, gz4, gz8, 0);
    __builtin_amdgcn_s_wait_tensorcnt((short)0);
  }
  __syncthreads();
#else
  for (int i = tid; i < CIN * WPX; i += 256) {
    const int r = i >> 7, c = i & (WPX - 1);
    xtile[r * PITCH + c] = xrow[(size_t)r * (HPX * WPX) + c];
  }
  __syncthreads();
#endif

  // ---- GEMM1: H1 = relu(W1 @ X + b1) -> buf[o][col] -------------------------
  for (int to = 0; to < 16; ++to) {
    const int row = to * 16 + m16;
    const float2* wp = (const float2*)(w1 + row * CIN + klo);
    __builtin_prefetch(w1 + row * CIN, 0, 1);   // global_prefetch_b8
    v8f acc = {};
    #pragma unroll 4
    for (int k = 0; k < CIN; k += 4) {
      float2 av = wp[k >> 1];
      v2f a; a.x = av.x; a.y = av.y;
      v2f b; b.x = xtile[(k + klo) * PITCH + col];
             b.y = xtile[(k + klo + 1) * PITCH + col];
      acc = wmma_f32(a, b, acc);
    }
    const int rb = to * 16 + (hi ? 8 : 0);      // D: VGPR r -> M = r (+8 hi half)
    #pragma unroll
    for (int r = 0; r < 8; ++r)
      buf[(rb + r) * PITCH + col] = fmaxf(acc[r] + b1[rb + r], 0.f);
  }
  __syncthreads();

  // ---- GEMM2: H2 = W2 @ H1 + b2 (accumulators in regs, then overwrite buf) --
  v8f acc2[16];
  for (int to = 0; to < 16; ++to) {
    const int row = to * 16 + m16;
    const float2* wp = (const float2*)(w2 + row * CIN + klo);
    __builtin_prefetch(w2 + row * CIN, 0, 1);
    v8f acc = {};
    #pragma unroll 4
    for (int k = 0; k < CIN; k += 4) {
      float2 av = wp[k >> 1];
      v2f a; a.x = av.x; a.y = av.y;
      v2f b; b.x = buf[(k + klo) * PITCH + col];
             b.y = buf[(k + klo + 1) * PITCH + col];
      acc = wmma_f32(a, b, acc);
    }
    acc2[to] = acc;
  }
  __syncthreads();
  for (int to = 0; to < 16; ++to) {
    const int rb = to * 16 + (hi ? 8 : 0);
    #pragma unroll
    for (int r = 0; r < 8; ++r)
      buf[(rb + r) * PITCH + col] = acc2[to][r] + b2[rb + r];
  }
  __syncthreads();

  // ---- faithful quirk: l2-normalize along W per (b,o,h) ----------------------
  {
    const int o = tid;                    // 256 threads = 256 channel rows
    float ss = 0.f;
    #pragma unroll 4
    for (int j = 0; j < WPX; ++j) { float v = buf[o * PITCH + j]; ss += v * v; }
    const float inv = 1.f / fmaxf(sqrtf(ss), 1e-12f);
    #pragma unroll 4
    for (int j = 0; j < WPX; ++j) buf[o * PITCH + j] *= inv;
  }
  __syncthreads();

  // ---- per-pixel LayerNorm(256) in place + defer l2 scale into pinv ----------
  if (tid < WPX) {
    const int p = tid;
    float s = 0.f, s2 = 0.f;
    for (int o = 0; o < CIN; ++o) {
      float v = buf[o * PITCH + p]; s += v; s2 += v * v;
    }
    const float mu   = s * (1.f / CIN);
    const float var  = fmaxf(s2 * (1.f / CIN) - mu * mu, 0.f);
    const float rstd = rsqrtf(var + 1e-5f);
    float ssy = 0.f;
    for (int o = 0; o < CIN; ++o) {
      float y = (buf[o * PITCH + p] - mu) * rstd * ln1g[o] + ln1b[o];
      buf[o * PITCH + p] = y;
      ssy += y * y;
    }
    pinv[p] = 1.f / fmaxf(sqrtf(ssy), 1e-12f);
  }
  __syncthreads();

  // ---- GEMM3: masks[px][n] = c_norm . protoN (K=256, N=192) ------------------
  v8f accm[12];
  for (int tn = 0; tn < 12; ++tn) {
    const int n = tn * 16 + m16;
    const float2* pp = (const float2*)(protosN + n * CIN + klo);
    __builtin_prefetch(protosN + n * CIN, 0, 1);
    v8f acc = {};
    #pragma unroll 4
    for (int k = 0; k < CIN; k += 4) {
      v2f a; a.x = buf[(k + klo) * PITCH + col];       // A: M=pixel, from LDS
             a.y = buf[(k + klo + 1) * PITCH + col];
      float2 bv = pp[k >> 1];
      v2f b; b.x = bv.x; b.y = bv.y;
      acc = wmma_f32(a, b, acc);
    }
    accm[tn] = acc;
  }
  __syncthreads();
  for (int tn = 0; tn < 12; ++tn) {
    #pragma unroll
    for (int r = 0; r < 8; ++r) {
      const int px = px0 + r + (hi ? 8 : 0);
      buf[px * MPITCH + tn * 16 + m16] = accm[tn][r] * pinv[px];
    }
  }
  __syncthreads();

  // ---- max over 10 prototypes, LayerNorm(19), store pre-resize seg map -------
  if (tid < WPX) {
    const int p = tid;
    float s19[NCLS];
    float s = 0.f;
    #pragma unroll
    for (int kc = 0; kc < NCLS; ++kc) {
      float mx = buf[p * MPITCH + kc * NPROTO];
      #pragma unroll
      for (int m = 1; m < NPROTO; ++m)
        mx = fmaxf(mx, buf[p * MPITCH + kc * NPROTO + m]);
      s19[kc] = mx; s += mx;
    }
    const float mu = s * (1.f / NCLS);
    float v = 0.f;
    #pragma unroll
    for (int kc = 0; kc < NCLS; ++kc) { float d = s19[kc] - mu; v += d * d; }
    const float rstd = rsqrtf(v * (1.f / NCLS) + 1e-5f);
    #pragma unroll
    for (int kc = 0; kc < NCLS; ++kc) {
      float y = (s19[kc] - mu) * rstd * ln2g[kc] + ln2b[kc];
      seg[(((size_t)bb * NCLS + kc) * HPX + hh) * WPX + p] = y;
    }
  }
}

// ---------------------------------------------------------------------------
// Kernel 3: 4x bicubic upsample 128x128 -> 512x512 (Keys a=-0.5, half-pixel)
// ---------------------------------------------------------------------------
__device__ __forceinline__ float cubw(float d) {
  d = fabsf(d);
  if (d <= 1.f) return (1.5f * d - 2.5f) * d * d + 1.f;
  if (d < 2.f)  return ((-0.5f * d + 2.5f) * d - 4.f) * d + 2.f;
  return 0.f;
}

__global__ void __launch_bounds__(256)
resize_bicubic(const float* __restrict__ seg, float* __restrict__ out) {
  const long long idx = (long long)blockIdx.x * blockDim.x + threadIdx.x;
  const long long total = (long long)BATCH * NCLS * OUTHW * OUTHW;
  if (idx >= total) return;
  const int ox = (int)(idx & (OUTHW - 1));
  const int oy = (int)((idx >> 9) & (OUTHW - 1));
  const int pl = (int)(idx >> 18);      // b*19 + k plane
  const float sx = (ox + 0.5f) * 0.25f - 0.5f;
  const float sy = (oy + 0.5f) * 0.25f - 0.5f;
  const int ix = (int)floorf(sx);
  const int iy = (int)floorf(sy);
  const float tx = sx - ix, ty = sy - iy;
  float wx[4], wy[4];
  wx[0] = cubw(tx + 1.f); wx[1] = cubw(tx);
  wx[2] = cubw(1.f - tx); wx[3] = cubw(2.f - tx);
  wy[0] = cubw(ty + 1.f); wy[1] = cubw(ty);
  wy[2] = cubw(1.f - ty); wy[3] = cubw(2.f - ty);
  const float* src = seg + (size_t)pl * (HPX * WPX);
  float acc = 0.f;
  #pragma unroll
  for (int j = 0; j < 4; ++j) {
    const int yy = min(max(iy - 1 + j, 0), HPX - 1);
    float rowv = 0.f;
    #pragma unroll
    for (int i = 0; i < 4; ++i) {
      const int xx = min(max(ix - 1 + i, 0), WPX - 1);
      rowv += wx[i] * src[yy * WPX + xx];
    }
    acc += wy[j] * rowv;
  }
  out[idx] = acc;
}

// ---------------------------------------------------------------------------
extern "C" void kernel_launch(void* const* d_in, const int* in_sizes, int n_in,
                              void* d_out, int out_size, void* d_ws,
                              size_t ws_size, hipStream_t stream) {
  (void)in_sizes; (void)n_in; (void)out_size; (void)ws_size;
  const float* x    = (const float*)d_in[0];
  const float* w1   = (const float*)d_in[1];
  const float* b1   = (const float*)d_in[2];
  const float* w2   = (const float*)d_in[3];
  const float* b2   = (const float*)d_in[4];
  const float* ln1g = (const float*)d_in[5];
  const float* ln1b = (const float*)d_in[6];
  const float* ln2g = (const float*)d_in[7];
  const float* ln2b = (const float*)d_in[8];
  const float* prot = (const float*)d_in[9];
  float* out = (float*)d_out;

  float* protosN = (float*)d_ws;            // 192*256 f32
  float* seg     = protosN + NPAD * CIN;    // 8*19*128*128 f32

  const size_t lds_bytes = (size_t)LDS_DW * 4;   // ~271 KB (<320 KB/WGP)
  (void)hipFuncSetAttribute((const void*)fused_head,
                            hipFuncAttributeMaxDynamicSharedMemorySize,
                            (int)lds_bytes);

  proto_l2norm<<<1, NPAD, 0, stream>>>(prot, protosN);
  fused_head<<<BATCH * HPX, 256, lds_bytes, stream>>>(
      x, w1, b1, w2, b2, ln1g, ln1b, ln2g, ln2b, protosN, seg);
  const long long total = (long long)BATCH * NCLS * OUTHW * OUTHW;
  resize_bicubic<<<(int)((total + 255) / 256), 256, 0, stream>>>(seg, out);
}